// ScaledDotAttention_39659728011691
// MI455X (gfx1250) — compile-verified
//
#include <hip/hip_runtime.h>

#define BATCH 4
#define SEQ   4096
#define DIM   256
#define NP    (BATCH * SEQ * DIM)   // elements per projected matrix (q/k/v)

typedef __attribute__((ext_vector_type(2))) float v2f;
typedef __attribute__((ext_vector_type(8))) float v8f;

// D = A(16x4,f32) * B(4x16,f32) + C(16x16,f32) -- exact fp32 matrix op on CDNA5
__device__ __forceinline__ v8f wmma4(v2f a, v2f b, v8f c) {
  return __builtin_amdgcn_wmma_f32_16x16x4_f32(false, a, false, b, (short)0, c,
                                               false, false);
}

// Order-preserving float<->uint encoding for atomicMax on floats
__device__ __forceinline__ unsigned enc_f(float f) {
  unsigned u = __float_as_uint(f);
  return (u & 0x80000000u) ? ~u : (u | 0x80000000u);
}
__device__ __forceinline__ float dec_f(unsigned u) {
  unsigned b = (u & 0x80000000u) ? (u ^ 0x80000000u) : ~u;
  return __uint_as_float(b);
}

__global__ void init_max_kernel(unsigned* slot) {
  *slot = 0x007FFFFFu;  // enc_f(-inf)
}

// ---------------------------------------------------------------------------
// Projections: O = X @ W, X:[B*S, 256], W:[256,256]. blockIdx.y picks q/k/v.
// Block = 256 thr (8 waves). Block computes a 16x256 strip; wave w -> 16x32.
// ---------------------------------------------------------------------------
__global__ __launch_bounds__(256) void proj_kernel(
    const float* __restrict__ Xq, const float* __restrict__ Xk,
    const float* __restrict__ Xv, const float* __restrict__ Wq,
    const float* __restrict__ Wk, const float* __restrict__ Wv,
    float* __restrict__ Oq, float* __restrict__ Ok, float* __restrict__ Ov) {
  __shared__ float As[16 * 260];  // stride 260: 16B-aligned, conflict-free
  const int z = blockIdx.y;
  const float* __restrict__ X = (z == 0) ? Xq : (z == 1) ? Xk : Xv;
  const float* __restrict__ W = (z == 0) ? Wq : (z == 1) ? Wk : Wv;
  float* __restrict__ O = (z == 0) ? Oq : (z == 1) ? Ok : Ov;
  const int m0 = blockIdx.x << 4;
  const int tid = threadIdx.x;
  {  // cooperative 16x256 A-panel load (coalesced float4)
    const int r = tid >> 4, c0 = (tid & 15) << 4;
    const float4* src =
        reinterpret_cast<const float4*>(X + (size_t)(m0 + r) * DIM + c0);
    float4* dst = reinterpret_cast<float4*>(&As[r * 260 + c0]);
#pragma unroll
    for (int j = 0; j < 4; ++j) dst[j] = src[j];
  }
  __syncthreads();
  const int lane = tid & 31, w = tid >> 5;
  const int lr = lane & 15, lh = lane >> 4;
  const int na = (w << 5) + lr, nb = na + 16;
  v8f c0 = {}, c1 = {};
#pragma unroll 4
  for (int t = 0; t < 64; ++t) {
    const int col = (t << 2) + (lh << 1);
    v2f a;
    a.x = As[lr * 260 + col];
    a.y = As[lr * 260 + col + 1];
    const float* wr = W + (size_t)col * DIM;
    v2f b0, b1;
    b0.x = wr[na];
    b0.y = wr[DIM + na];
    b1.x = wr[nb];
    b1.y = wr[DIM + nb];
    c0 = wmma4(a, b0, c0);
    c1 = wmma4(a, b1, c1);
  }
#pragma unroll
  for (int r = 0; r < 8; ++r) {
    const int row = m0 + r + (lh << 3);
    O[(size_t)row * DIM + na] = c0[r];
    O[(size_t)row * DIM + nb] = c1[r];
  }
}

// ---------------------------------------------------------------------------
// Scores: S[b,i,j] = dot(q[b,i], k[b,j]) / 16, written raw to attn region,
// plus global max reduction (pre-bias, post-scale, matching reference).
// Block computes a 64x64 tile; wave (wm=w>>1, wn=w&1) -> 16x32. K chunked x64.
// ---------------------------------------------------------------------------
__global__ __launch_bounds__(256) void scores_kernel(
    const float* __restrict__ qw, const float* __restrict__ kw,
    float* __restrict__ attn, unsigned* __restrict__ maxslot) {
  __shared__ float Qs[64 * 68];
  __shared__ float Ks[64 * 68];
  __shared__ float wred[8];
  const int b = blockIdx.z;
  const int j0 = blockIdx.x << 6;
  const int i0 = blockIdx.y << 6;
  const float* __restrict__ Qb = qw + (size_t)b * SEQ * DIM;
  const float* __restrict__ Kb = kw + (size_t)b * SEQ * DIM;
  const int tid = threadIdx.x;
  const int lane = tid & 31, w = tid >> 5;
  const int lr = lane & 15, lh = lane >> 4;
  const int wm = w >> 1, wn = w & 1;
  const int r = tid >> 2, cb = (tid & 3) << 4;
  v8f c0 = {}, c1 = {};
  for (int kc = 0; kc < 4; ++kc) {
    {
      const float4* qs = reinterpret_cast<const float4*>(
          Qb + (size_t)(i0 + r) * DIM + (kc << 6) + cb);
      const float4* ks = reinterpret_cast<const float4*>(
          Kb + (size_t)(j0 + r) * DIM + (kc << 6) + cb);
      float4* qd = reinterpret_cast<float4*>(&Qs[r * 68 + cb]);
      float4* kd = reinterpret_cast<float4*>(&Ks[r * 68 + cb]);
#pragma unroll
      for (int j = 0; j < 4; ++j) {
        qd[j] = qs[j];
        kd[j] = ks[j];
      }
    }
    __syncthreads();
    const float* qrow = &Qs[(wm * 16 + lr) * 68];
    const float* k0 = &Ks[((wn << 5) + lr) * 68];
    const float* k1 = &Ks[((wn << 5) + 16 + lr) * 68];
#pragma unroll
    for (int t = 0; t < 16; ++t) {
      const int col = (t << 2) + (lh << 1);
      v2f a, b0, b1;
      a.x = qrow[col];
      a.y = qrow[col + 1];
      b0.x = k0[col];
      b0.y = k0[col + 1];
      b1.x = k1[col];
      b1.y = k1[col + 1];
      c0 = wmma4(a, b0, c0);
      c1 = wmma4(a, b1, c1);
    }
    __syncthreads();
  }
  // scale by 1/sqrt(256) and reduce block max
  float m = -3.402823466e38f;
#pragma unroll
  for (int rr = 0; rr < 8; ++rr) {
    c0[rr] *= 0.0625f;
    c1[rr] *= 0.0625f;
    m = fmaxf(m, fmaxf(c0[rr], c1[rr]));
  }
#pragma unroll
  for (int off = 16; off > 0; off >>= 1) m = fmaxf(m, __shfl_xor(m, off, 32));
  if (lane == 0) wred[w] = m;
  __syncthreads();
  if (tid == 0) {
    float mm = wred[0];
#pragma unroll
    for (int i = 1; i < 8; ++i) mm = fmaxf(mm, wred[i]);
    atomicMax(maxslot, enc_f(mm));
  }
  float* arow = attn + (size_t)b * SEQ * SEQ;
#pragma unroll
  for (int rr = 0; rr < 8; ++rr) {
    const int row = i0 + wm * 16 + rr + (lh << 3);
    arow[(size_t)row * SEQ + j0 + (wn << 5) + lr] = c0[rr];
    arow[(size_t)row * SEQ + j0 + (wn << 5) + 16 + lr] = c1[rr];
  }
}

// ---------------------------------------------------------------------------
// Softmax per row with diagonal bias 0.5*globalMax, in place on attn region.
// One block (256 thr) per row of 4096; 16 elements per thread in registers.
// ---------------------------------------------------------------------------
__global__ __launch_bounds__(256) void softmax_kernel(
    float* __restrict__ attn, const unsigned* __restrict__ maxslot) {
  __shared__ float wred[8];
  __shared__ float bval;
  const int rowg = blockIdx.x;  // b*S + i
  const int i = rowg & (SEQ - 1);
  float* __restrict__ p = attn + (size_t)rowg * SEQ;
  const float Mg = dec_f(*maxslot) * 0.5f;
  const int tid = threadIdx.x;
  const int lane = tid & 31, w = tid >> 5;
  float v[16];
  float rmax = -3.402823466e38f;
#pragma unroll
  for (int u = 0; u < 4; ++u) {
    const int idx4 = (u << 8) + tid;
    float4 t = reinterpret_cast<const float4*>(p)[idx4];
    float* tf = reinterpret_cast<float*>(&t);
    const int base = idx4 << 2;
    if ((unsigned)(i - base) < 4u) tf[i - base] += Mg;  // diagonal bias
#pragma unroll
    for (int c = 0; c < 4; ++c) {
      v[u * 4 + c] = tf[c];
      rmax = fmaxf(rmax, tf[c]);
    }
  }
#pragma unroll
  for (int off = 16; off > 0; off >>= 1)
    rmax = fmaxf(rmax, __shfl_xor(rmax, off, 32));
  if (lane == 0) wred[w] = rmax;
  __syncthreads();
  if (tid == 0) {
    float mm = wred[0];
    for (int k = 1; k < 8; ++k) mm = fmaxf(mm, wred[k]);
    bval = mm;
  }
  __syncthreads();
  rmax = bval;
  float s = 0.f;
#pragma unroll
  for (int e = 0; e < 16; ++e) {
    v[e] = expf(v[e] - rmax);
    s += v[e];
  }
#pragma unroll
  for (int off = 16; off > 0; off >>= 1) s += __shfl_xor(s, off, 32);
  if (lane == 0) wred[w] = s;
  __syncthreads();
  if (tid == 0) {
    float ss = wred[0];
    for (int k = 1; k < 8; ++k) ss += wred[k];
    bval = ss;
  }
  __syncthreads();
  const float inv = 1.0f / bval;
#pragma unroll
  for (int u = 0; u < 4; ++u) {
    float4 t;
    float* tf = reinterpret_cast<float*>(&t);
#pragma unroll
    for (int c = 0; c < 4; ++c) tf[c] = v[u * 4 + c] * inv;
    reinterpret_cast<float4*>(p)[(u << 8) + tid] = t;
  }
}

// ---------------------------------------------------------------------------
// Output: out[b] = attn[b] (4096x4096) @ v[b] (4096x256). Block = 64 rows x
// 256 cols (amortizes V reads). Wave w: rows (w>>1)*16, cols (w&1)*128 ->
// 8 accumulator tiles per wave. A staged in LDS per 64-wide K chunk.
// ---------------------------------------------------------------------------
__global__ __launch_bounds__(256) void out_kernel(
    const float* __restrict__ attn, const float* __restrict__ vw,
    float* __restrict__ out) {
  __shared__ float As[64 * 68];
  const int m0 = blockIdx.x << 6;  // global row = b*S + i
  const int b = m0 >> 12;
  const float* __restrict__ V = vw + (size_t)b * SEQ * DIM;
  const int tid = threadIdx.x;
  const int lane = tid & 31, w = tid >> 5;
  const int lr = lane & 15, lh = lane >> 4;
  const int wm = w >> 1;
  const int wn = (w & 1) << 7;
  const int r = tid >> 2, cb = (tid & 3) << 4;
  v8f acc[8] = {};
  for (int kc = 0; kc < 64; ++kc) {
    {
      const float4* src = reinterpret_cast<const float4*>(
          attn + (size_t)(m0 + r) * SEQ + (kc << 6) + cb);
      float4* dst = reinterpret_cast<float4*>(&As[r * 68 + cb]);
#pragma unroll
      for (int j = 0; j < 4; ++j) dst[j] = src[j];
      if (kc + 1 < 64)  // gfx1250 global_prefetch of next A chunk
        __builtin_prefetch(attn + (size_t)(m0 + r) * SEQ + ((kc + 1) << 6) + cb,
                           0, 1);
    }
    __syncthreads();
    const float* arow = &As[(wm * 16 + lr) * 68];
#pragma unroll 4
    for (int t = 0; t < 16; ++t) {
      const int col = (t << 2) + (lh << 1);
      v2f a;
      a.x = arow[col];
      a.y = arow[col + 1];
      const int k4 = (kc << 6) + col;
      const float* vr = V + (size_t)k4 * DIM;
#pragma unroll
      for (int tt = 0; tt < 8; ++tt) {
        const int n = wn + (tt << 4) + lr;
        v2f bv;
        bv.x = vr[n];
        bv.y = vr[DIM + n];
        acc[tt] = wmma4(a, bv, acc[tt]);
      }
    }
    __syncthreads();
  }
#pragma unroll
  for (int tt = 0; tt < 8; ++tt) {
#pragma unroll
    for (int rr = 0; rr < 8; ++rr) {
      const int row = m0 + wm * 16 + rr + (lh << 3);
      out[(size_t)row * DIM + wn + (tt << 4) + lr] = acc[tt][rr];
    }
  }
}

extern "C" void kernel_launch(void* const* d_in, const int* in_sizes, int n_in,
                              void* d_out, int out_size, void* d_ws,
                              size_t ws_size, hipStream_t stream) {
  const float* Q = (const float*)d_in[0];
  const float* K = (const float*)d_in[1];
  const float* V = (const float*)d_in[2];
  const float* Wq = (const float*)d_in[3];
  const float* Wk = (const float*)d_in[4];
  const float* Wv = (const float*)d_in[5];

  float* out = (float*)d_out;                       // [B,S,D]
  float* attn = out + (size_t)BATCH * SEQ * DIM;    // [B,S,S]

  float* qw = (float*)d_ws;                         // 16 MB
  float* kw = qw + (size_t)NP;                      // 16 MB
  float* vw = kw + (size_t)NP;                      // 16 MB
  unsigned* maxslot = (unsigned*)(vw + (size_t)NP); // 4 B

  init_max_kernel<<<1, 1, 0, stream>>>(maxslot);
  proj_kernel<<<dim3((BATCH * SEQ) / 16, 3), 256, 0, stream>>>(
      Q, K, V, Wq, Wk, Wv, qw, kw, vw);
  scores_kernel<<<dim3(SEQ / 64, SEQ / 64, BATCH), 256, 0, stream>>>(
      qw, kw, attn, maxslot);
  softmax_kernel<<<BATCH * SEQ, 256, 0, stream>>>(attn, maxslot);
  out_kernel<<<(BATCH * SEQ) / 64, 256, 0, stream>>>(attn, vw, out);
}